// EntityClassify_26577257628123
// MI455X (gfx1250) — compile-verified
//
#include <hip/hip_runtime.h>
#include <hip/hip_bf16.h>
#include <stdint.h>

#define NP  200000
#define NA  100000
#define NE  500000
#define DIN 128
#define DOUT 16

typedef __attribute__((ext_vector_type(2))) float v2f;
typedef __attribute__((ext_vector_type(8))) float v8f;

// ---------------------------------------------------------------------------
// Y[nrows x NOUT] = X[nrows x 128] @ Wm[128 x NOUT] + bias, f32 WMMA 16x16x4.
// Block = 256 threads = 8 waves; each wave owns a 16-row M tile (128 rows/blk).
// Wave32 fragment layout per CDNA5 ISA:
//   A 16x4 : lane half h, lane r: a[v] = X[r][k0 + 2h + v]
//   B 4x16 : b[v] = W[k0 + 2h + v][n0 + r]
//   C/D    : acc[v] = element (M = v + 8h, N = r)
// ---------------------------------------------------------------------------
template <int NOUT>
__global__ __launch_bounds__(256) void gemm_bias_wmma(
    const float* __restrict__ X, const float* __restrict__ Wm,
    const float* __restrict__ bias, float* __restrict__ Y, int nrows)
{
    __shared__ float Wlds[DIN * NOUT];
    for (int i = threadIdx.x; i < DIN * NOUT; i += 256) Wlds[i] = Wm[i];
    __syncthreads();

    const int lane = threadIdx.x & 31;
    const int wave = threadIdx.x >> 5;
    const int h    = lane >> 4;    // lane half (0 or 1)
    const int r    = lane & 15;
    const int rowBase = blockIdx.x * 128 + wave * 16;

    int loadRow = rowBase + r;                 // clamp loads so EXEC stays all-1
    if (loadRow >= nrows) loadRow = nrows - 1;
    const float* xrow = X + (int64_t)loadRow * DIN;

    const bool fullTile = (rowBase + 16 <= nrows);

    for (int nt = 0; nt < NOUT / 16; ++nt) {
        const int n0 = nt * 16;
        v8f acc = {};
#pragma unroll 4
        for (int k0 = 0; k0 < DIN; k0 += 4) {
            v2f a, b;
            a.x = xrow[k0 + 2 * h];
            a.y = xrow[k0 + 2 * h + 1];
            b.x = Wlds[(k0 + 2 * h) * NOUT + n0 + r];
            b.y = Wlds[(k0 + 2 * h + 1) * NOUT + n0 + r];
            acc = __builtin_amdgcn_wmma_f32_16x16x4_f32(
                false, a, false, b, (short)0, acc, false, false);
        }
        const float bv = bias[n0 + r];
        if (fullTile) {
            // hot path: no per-element guards -> plain global_store_b32 chain
#pragma unroll
            for (int v = 0; v < 8; ++v)
                Y[(int64_t)(rowBase + v + 8 * h) * NOUT + n0 + r] = acc[v] + bv;
        } else {
#pragma unroll
            for (int v = 0; v < 8; ++v) {
                const int row = rowBase + v + 8 * h;
                if (row < nrows)
                    Y[(int64_t)row * NOUT + n0 + r] = acc[v] + bv;
            }
        }
    }
}

// ---------------------------------------------------------------------------
// Edge scatter: agg[dst[e]][c..c+3] += m[src[e]][c..c+3].
// One thread per (edge, 4-col group): coalesced b128 gather, 4x f32 atomics.
// Agg buffers are L2-resident (<=102 MB vs 192 MB L2) so atomics stay in L2.
// LOGG = log2(D/4) groups per edge; D = 4 << LOGG.
// ---------------------------------------------------------------------------
template <int LOGG>
__global__ __launch_bounds__(256) void scatter_add4_kernel(
    const float* __restrict__ m, const int* __restrict__ srci,
    const int* __restrict__ dsti, float* __restrict__ agg, int nedges)
{
    const int64_t i = (int64_t)blockIdx.x * blockDim.x + threadIdx.x;
    const int64_t total = (int64_t)nedges << LOGG;
    if (i >= total) return;
    const int e = (int)(i >> LOGG);
    const int g = (int)(i & ((1 << LOGG) - 1));
    const int s = srci[e];
    const int d = dsti[e];
    const float4 v = ((const float4*)m)[(((int64_t)s) << LOGG) + g];
    float* p = agg + (((int64_t)d) << (LOGG + 2)) + g * 4;
    atomicAdd(p + 0, v.x);
    atomicAdd(p + 1, v.y);
    atomicAdd(p + 2, v.z);
    atomicAdd(p + 3, v.w);
}

__global__ __launch_bounds__(256) void zero4_kernel(float4* __restrict__ p, int64_t n4)
{
    const int64_t i = (int64_t)blockIdx.x * blockDim.x + threadIdx.x;
    if (i < n4) p[i] = make_float4(0.f, 0.f, 0.f, 0.f);
}

// y = relu(relu(a) + b)   (per-relation ReLU order: first rel, then second)
__global__ __launch_bounds__(256) void relu_combine2_kernel(
    const float4* __restrict__ a, const float4* __restrict__ b,
    float4* __restrict__ y, int64_t n4)
{
    const int64_t i = (int64_t)blockIdx.x * blockDim.x + threadIdx.x;
    if (i < n4) {
        const float4 av = a[i];
        const float4 bv = b[i];
        float4 o;
        o.x = fmaxf(fmaxf(av.x, 0.f) + bv.x, 0.f);
        o.y = fmaxf(fmaxf(av.y, 0.f) + bv.y, 0.f);
        o.z = fmaxf(fmaxf(av.z, 0.f) + bv.z, 0.f);
        o.w = fmaxf(fmaxf(av.w, 0.f) + bv.w, 0.f);
        y[i] = o;
    }
}

// y = relu(a)
__global__ __launch_bounds__(256) void relu1_kernel(
    const float4* __restrict__ a, float4* __restrict__ y, int64_t n4)
{
    const int64_t i = (int64_t)blockIdx.x * blockDim.x + threadIdx.x;
    if (i < n4) {
        const float4 av = a[i];
        y[i] = make_float4(fmaxf(av.x, 0.f), fmaxf(av.y, 0.f),
                           fmaxf(av.z, 0.f), fmaxf(av.w, 0.f));
    }
}

static inline int blks(int64_t n, int t) { return (int)((n + t - 1) / t); }

extern "C" void kernel_launch(void* const* d_in, const int* in_sizes, int n_in,
                              void* d_out, int out_size, void* d_ws, size_t ws_size,
                              hipStream_t stream)
{
    const float* embed_paper  = (const float*)d_in[0];
    const float* embed_author = (const float*)d_in[1];
    const float* W1c  = (const float*)d_in[2];  const float* b1c  = (const float*)d_in[3];
    const float* W1wb = (const float*)d_in[4];  const float* b1wb = (const float*)d_in[5];
    const float* W1w  = (const float*)d_in[6];  const float* b1w  = (const float*)d_in[7];
    const float* W2c  = (const float*)d_in[8];  const float* b2c  = (const float*)d_in[9];
    const float* W2wb = (const float*)d_in[10]; const float* b2wb = (const float*)d_in[11];
    const float* W2w  = (const float*)d_in[12]; const float* b2w  = (const float*)d_in[13];
    const int* cites_src = (const int*)d_in[14];
    const int* cites_dst = (const int*)d_in[15];
    const int* wb_src    = (const int*)d_in[16];
    const int* wb_dst    = (const int*)d_in[17];
    const int* wr_src    = (const int*)d_in[18];
    const int* wr_dst    = (const int*)d_in[19];

    const int64_t szP = (int64_t)NP * DIN;   // 25.6M floats
    const int64_t szA = (int64_t)NA * DIN;   // 12.8M floats
    const int64_t sP2 = (int64_t)NP * DOUT;  // 3.2M floats
    const int64_t sA2 = (int64_t)NA * DOUT;  // 1.6M floats

    float* ws = (float*)d_ws;
    float* B0 = ws;              // m_paper(cites)    -> later h1_paper
    float* B1 = B0 + szP;        // m_paper(writtenby)-> later layer-2 m buffers
    float* B2 = B1 + szP;        // m_author(writes)  -> later h1_author
    float* B3 = B2 + szA;        // agg_paper(cites)  -> later layer-2 aggs
    float* B4 = B3 + szP;        // agg_paper(writes)
    float* B5 = B4 + szP;        // agg_author(writtenby)

    // ---------------- Layer 1: dense transforms (f32 WMMA) ----------------
    gemm_bias_wmma<128><<<(NP + 127) / 128, 256, 0, stream>>>(embed_paper,  W1c,  b1c,  B0, NP);
    gemm_bias_wmma<128><<<(NP + 127) / 128, 256, 0, stream>>>(embed_paper,  W1wb, b1wb, B1, NP);
    gemm_bias_wmma<128><<<(NA + 127) / 128, 256, 0, stream>>>(embed_author, W1w,  b1w,  B2, NA);

    // B3,B4,B5 are contiguous: zero them in one vectorized launch
    const int64_t zero1_n4 = (2 * szP + szA) / 4;
    zero4_kernel<<<blks(zero1_n4, 256), 256, 0, stream>>>((float4*)B3, zero1_n4);

    // ---------------- Layer 1: edge scatter-sum (D=128 -> 32 groups) -------
    const int64_t e128g = (int64_t)NE << 5;
    scatter_add4_kernel<5><<<blks(e128g, 256), 256, 0, stream>>>(B0, cites_src, cites_dst, B3, NE);
    scatter_add4_kernel<5><<<blks(e128g, 256), 256, 0, stream>>>(B2, wr_src,    wr_dst,    B4, NE);
    scatter_add4_kernel<5><<<blks(e128g, 256), 256, 0, stream>>>(B1, wb_src,    wb_dst,    B5, NE);

    // h1_paper = relu(relu(agg_cites) + agg_writes); h1_author = relu(agg_writtenby)
    relu_combine2_kernel<<<blks(szP / 4, 256), 256, 0, stream>>>(
        (const float4*)B3, (const float4*)B4, (float4*)B0, szP / 4);
    relu1_kernel<<<blks(szA / 4, 256), 256, 0, stream>>>(
        (const float4*)B5, (float4*)B2, szA / 4);

    // ---------------- Layer 2 (reuse B1 / B3 space) ----------------
    float* M2c  = B1;
    float* M2wb = B1 + sP2;
    float* M2w  = B1 + 2 * sP2;
    float* A2c  = B3;            // agg_paper(cites)
    float* A2w  = B3 + sP2;      // agg_paper(writes)
    float* A2wb = B3 + 2 * sP2;  // agg_author(writtenby)

    gemm_bias_wmma<16><<<(NP + 127) / 128, 256, 0, stream>>>(B0, W2c,  b2c,  M2c,  NP);
    gemm_bias_wmma<16><<<(NP + 127) / 128, 256, 0, stream>>>(B0, W2wb, b2wb, M2wb, NP);
    gemm_bias_wmma<16><<<(NA + 127) / 128, 256, 0, stream>>>(B2, W2w,  b2w,  M2w,  NA);

    // A2c,A2w,A2wb contiguous: one zero launch
    const int64_t zero2_n4 = (2 * sP2 + sA2) / 4;
    zero4_kernel<<<blks(zero2_n4, 256), 256, 0, stream>>>((float4*)A2c, zero2_n4);

    // ---------------- Layer 2: edge scatter-sum (D=16 -> 4 groups) ---------
    const int64_t e16g = (int64_t)NE << 2;
    scatter_add4_kernel<2><<<blks(e16g, 256), 256, 0, stream>>>(M2c,  cites_src, cites_dst, A2c,  NE);
    scatter_add4_kernel<2><<<blks(e16g, 256), 256, 0, stream>>>(M2w,  wr_src,    wr_dst,    A2w,  NE);
    scatter_add4_kernel<2><<<blks(e16g, 256), 256, 0, stream>>>(M2wb, wb_src,    wb_dst,    A2wb, NE);

    // ------------- Final: out = [paper(200000x16), author(100000x16)] ------
    float* outP = (float*)d_out;
    float* outA = outP + sP2;
    relu_combine2_kernel<<<blks(sP2 / 4, 256), 256, 0, stream>>>(
        (const float4*)A2c, (const float4*)A2w, (float4*)outP, sP2 / 4);
    relu1_kernel<<<blks(sA2 / 4, 256), 256, 0, stream>>>(
        (const float4*)A2wb, (float4*)outA, sA2 / 4);
}